// GVPInputFeaturizer_77438260347480
// MI455X (gfx1250) — compile-verified
//
#include <hip/hip_runtime.h>
#include <math.h>

// Problem shape (fixed by setup_inputs).
#define BB 8
#define NN 2048
#define NT3 (3 * NN)

typedef __attribute__((ext_vector_type(2))) float v2f;
typedef __attribute__((ext_vector_type(8))) float v8f;

// Fast HW math (raw v_sqrt_f32 / v_rsq_f32, no correction sequence).
#if __has_builtin(__builtin_amdgcn_sqrtf)
#define FSQRT(x) __builtin_amdgcn_sqrtf(x)
#else
#define FSQRT(x) sqrtf(x)
#endif
#if __has_builtin(__builtin_amdgcn_rsqf)
#define FRSQ(x) __builtin_amdgcn_rsqf(x)
#else
#define FRSQ(x) (1.0f / sqrtf(x))
#endif

// gfx1250 async global->LDS staging (ASYNCcnt-tracked), guarded.
// Probe-discovered signature: (global int*, shared int*, imm offset, imm cpol).
#if __has_builtin(__builtin_amdgcn_global_load_async_to_lds_b32) && \
    __has_builtin(__builtin_amdgcn_s_wait_asynccnt)
#define HAVE_ASYNC_LDS 1
typedef __attribute__((address_space(1))) int GI;    // global int
typedef __attribute__((address_space(3))) int LI;    // LDS int
#endif

// ---------------- small float3 helpers ----------------
struct F3 { float x, y, z; };
__device__ __forceinline__ F3 f3sub(F3 a, F3 b) { return {a.x - b.x, a.y - b.y, a.z - b.z}; }
__device__ __forceinline__ F3 f3add(F3 a, F3 b) { return {a.x + b.x, a.y + b.y, a.z + b.z}; }
__device__ __forceinline__ float f3dot(F3 a, F3 b) { return a.x * b.x + a.y * b.y + a.z * b.z; }
__device__ __forceinline__ F3 f3cross(F3 a, F3 b) {
  return {a.y * b.z - a.z * b.y, a.z * b.x - a.x * b.z, a.x * b.y - a.y * b.x};
}
// reference _normalize: x / sqrt(sum(x^2) + 1e-8)
__device__ __forceinline__ F3 f3nrm(F3 v) {
  float inv = FRSQ(f3dot(v, v) + 1e-8f);
  return {v.x * inv, v.y * inv, v.z * inv};
}

// coords layout: (B, N, 3 atoms, 3 xyz)
__device__ __forceinline__ F3 ld_atom(const float* C, int b, int n, int ai) {
  const float* p = C + (((size_t)b * NN + n) * 3 + ai) * 3;
  return {p[0], p[1], p[2]};
}
__device__ __forceinline__ F3 ld_atom_flat(const float* C, int b, int aflat) {
  return ld_atom(C, b, aflat / 3, aflat % 3);
}

// ---------------- kernel 1: node scalar + vector features ----------------
__global__ void gvp_node_kernel(const float* __restrict__ C,
                                const unsigned char* __restrict__ cmask,
                                float* __restrict__ ns, float* __restrict__ nv) {
  int g = blockIdx.x * blockDim.x + threadIdx.x;
  if (g >= BB * NN) return;
  int b = g / NN, n = g % NN;

  // --- dihedrals: padded D index t = 3n + s; valid for 1 <= t <= 3N-3 ---
  float cd[3], sd[3];
#pragma unroll
  for (int s = 0; s < 3; ++s) {
    int t = 3 * n + s;
    float D = 0.0f;
    if (t >= 1 && t <= NT3 - 3) {
      int q = t - 1;  // uses backbone atoms q..q+3
      F3 p0 = ld_atom_flat(C, b, q);
      F3 p1 = ld_atom_flat(C, b, q + 1);
      F3 p2 = ld_atom_flat(C, b, q + 2);
      F3 p3 = ld_atom_flat(C, b, q + 3);
      F3 u2 = f3nrm(f3sub(p1, p0));
      F3 u1 = f3nrm(f3sub(p2, p1));
      F3 u0 = f3nrm(f3sub(p3, p2));
      F3 n2 = f3nrm(f3cross(u2, u1));
      F3 n1 = f3nrm(f3cross(u1, u0));
      float c = f3dot(n2, n1);
      c = fminf(fmaxf(c, -1.0f + 1e-7f), 1.0f - 1e-7f);
      float sg = f3dot(u2, n1);
      float sgn = (sg > 0.0f) ? 1.0f : ((sg < 0.0f) ? -1.0f : 0.0f);
      D = sgn * acosf(c);
    }
    cd[s] = cosf(D);
    sd[s] = sinf(D);
  }
  float* o = ns + (size_t)g * 7;
  o[0] = cd[0]; o[1] = cd[1]; o[2] = cd[2];
  o[3] = sd[0]; o[4] = sd[1]; o[5] = sd[2];
  o[6] = cmask[g] ? 1.0f : 0.0f;

  // --- orientations + sidechain (CA = atom 1) ---
  F3 xc = ld_atom(C, b, n, 1);
  F3 fwd = {0, 0, 0}, bwd = {0, 0, 0};
  if (n < NN - 1) fwd = f3nrm(f3sub(ld_atom(C, b, n + 1, 1), xc));
  if (n > 0)      bwd = f3nrm(f3sub(ld_atom(C, b, n - 1, 1), xc));

  F3 nn_ = f3nrm(f3sub(ld_atom(C, b, n, 0), xc));  // N - CA
  F3 cc  = f3nrm(f3sub(ld_atom(C, b, n, 2), xc));  // C - CA
  F3 bis = f3nrm(f3add(cc, nn_));
  F3 pp  = f3nrm(f3cross(cc, nn_));
  const float s13 = 0.5773502691896258f;   // sqrt(1/3)
  const float s23 = 0.8164965809277260f;   // sqrt(2/3)
  F3 side = {-bis.x * s13 - pp.x * s23,
             -bis.y * s13 - pp.y * s23,
             -bis.z * s13 - pp.z * s23};

  float* ov = nv + (size_t)g * 9;
  ov[0] = fwd.x;  ov[1] = fwd.y;  ov[2] = fwd.z;
  ov[3] = bwd.x;  ov[4] = bwd.y;  ov[5] = bwd.z;
  ov[6] = side.x; ov[7] = side.y; ov[8] = side.z;
}

// ---------------- kernel 2: WMMA pairwise distances + top-k ----------------
// One 256-thread block per 16-row strip. Dynamic LDS layout:
//   sX   : float4[NN]        (CA coords, w = 0 -> K=3 zero pad)    32 KB
//   sN2  : float [NN]        squared norms                           8 KB
//   sRes : int   [NN]        residue indices                         8 KB
//   sCM  : int   [NN]        coord mask (1/0)                        8 KB
//   sRM  : int   [NN]        !padding (1/0)                          8 KB
//   sD   : float [16][NN]    adjusted distances for the strip      128 KB
#define SM_X   0
#define SM_N2  (NN * 16)
#define SM_RES (SM_N2 + NN * 4)
#define SM_CM  (SM_RES + NN * 4)
#define SM_RM  (SM_CM + NN * 4)
#define SM_D   (SM_RM + NN * 4)
#define SM_TOT (SM_D + 16 * NN * 4)   // 196608 bytes

__global__ void gvp_dist_kernel(const float* __restrict__ C,
                                const unsigned char* __restrict__ cmask,
                                const int* __restrict__ ridx,
                                const unsigned char* __restrict__ pmask,
                                float* __restrict__ outD, float* __restrict__ outE,
                                float* __restrict__ outC, float* __restrict__ outR,
                                int K) {
  extern __shared__ __align__(16) char smem[];
  float4* sX  = (float4*)(smem + SM_X);
  float*  sN2 = (float*)(smem + SM_N2);
  int*    sRes = (int*)(smem + SM_RES);
  int*    sCM = (int*)(smem + SM_CM);
  int*    sRM = (int*)(smem + SM_RM);
  float*  sD  = (float*)(smem + SM_D);

  const int tid = threadIdx.x;
  const int blk = blockIdx.x;
  const int b  = blk >> 7;            // blk / (NN/16)
  const int r0 = (blk & 127) << 4;    // strip start row

  // ---- phase 1: stage batch-b CA data into LDS ----
#if defined(HAVE_ASYNC_LDS)
  // Async DMA the CA xyz + res_idx straight into LDS (no VGPR round trip),
  // tracked by ASYNCcnt. Masks convert via normal path.
  for (int n = tid; n < NN; n += 256) {
    const float* p = C + (((size_t)b * NN + n) * 3 + 1) * 3;
    __builtin_amdgcn_global_load_async_to_lds_b32(
        (GI*)(p + 0), (LI*)&sX[n].x, 0, 0);
    __builtin_amdgcn_global_load_async_to_lds_b32(
        (GI*)(p + 1), (LI*)&sX[n].y, 0, 0);
    __builtin_amdgcn_global_load_async_to_lds_b32(
        (GI*)(p + 2), (LI*)&sX[n].z, 0, 0);
    __builtin_amdgcn_global_load_async_to_lds_b32(
        (GI*)(ridx + (size_t)b * NN + n), (LI*)&sRes[n], 0, 0);
    sCM[n] = cmask[(size_t)b * NN + n] ? 1 : 0;
    sRM[n] = pmask[(size_t)b * NN + n] ? 0 : 1;
  }
  __builtin_amdgcn_s_wait_asynccnt(0);
  __syncthreads();
  for (int n = tid; n < NN; n += 256) {
    float x = sX[n].x, y = sX[n].y, z = sX[n].z;
    sX[n].w = 0.0f;                    // K=3 zero pad for the WMMA fragment
    sN2[n] = x * x + y * y + z * z;
  }
#else
  for (int n = tid; n < NN; n += 256) {
    const float* p = C + (((size_t)b * NN + n) * 3 + 1) * 3;
    float x = p[0], y = p[1], z = p[2];
    sX[n] = make_float4(x, y, z, 0.0f);
    sN2[n] = x * x + y * y + z * z;
    sRes[n] = ridx[(size_t)b * NN + n];
    sCM[n] = cmask[(size_t)b * NN + n] ? 1 : 0;
    sRM[n] = pmask[(size_t)b * NN + n] ? 0 : 1;
  }
#endif
  __syncthreads();

  const int lane = tid & 31;
  const int wv = tid >> 5;
  const int hi = lane >> 4;   // K-half select (A/B frags), M+8 select (D frag)
  const int m = lane & 15;

  // A fragment: rows r0..r0+15; lanes 0-15 carry K={0,1}, lanes 16-31 K={2,3}.
  float4 xa = sX[r0 + m];
  v2f afrag;
  afrag[0] = hi ? xa.z : xa.x;
  afrag[1] = hi ? xa.w : xa.y;   // xa.w == 0 -> K=3 pad

  // Hoist per-row strip data (row for acc[v] is r0 + v + 8*hi).
  float ni[8], cmiF[8], rmiF[8];
  int rsi[8], ivv[8];
#pragma unroll
  for (int v = 0; v < 8; ++v) {
    int iv = r0 + v + 8 * hi;
    ivv[v] = iv;
    ni[v] = sN2[iv];
    cmiF[v] = (float)sCM[iv];
    rsi[v] = sRes[iv];
    rmiF[v] = (float)sRM[iv];
  }

  // ---- phase 2: WMMA gram tiles -> fused D_adjust strip in LDS ----
  for (int ct = wv; ct < NN / 16; ct += 8) {
    int c0 = ct << 4;
    int j = c0 + m;
    float4 xb = sX[j];
    v2f bfrag;
    bfrag[0] = hi ? xb.z : xb.x;
    bfrag[1] = hi ? xb.w : xb.y;

    v8f acc = {};
    // D[16x16] = A(16x4) * B(4x16): acc[v] = dot(x_{r0+v+8*hi}, x_{c0+m})
    acc = __builtin_amdgcn_wmma_f32_16x16x4_f32(
        /*neg_a=*/false, afrag, /*neg_b=*/false, bfrag,
        /*c_mod=*/(short)0, acc, /*reuse_a=*/false, /*reuse_b=*/false);

    float nj = sN2[j];
    float cmjF = (float)sCM[j], rmjF = (float)sRM[j];
    int rsj = sRes[j];
#pragma unroll
    for (int v = 0; v < 8; ++v) {
      int iv = ivv[v];
      float d2 = ni[v] + nj - 2.0f * acc[v];
      float c2 = cmiF[v] * cmjF;                     // 1 iff both coords valid
      float Dv = FSQRT(fmaxf(d2, 0.0f) + 1e-8f) * c2;
      int dr = rsi[v] - rsj; if (dr < 0) dr = -dr;
      float covf = (dr <= 3) ? 0.0f : 1.0f;          // covalent exclusion
      float seqd = fabsf((float)(iv - j));
      // branch-free D_adjust: D*(1-cov) + !cm2*(1e8+seq*1e6) + !rm2*1e10
      float Da = Dv * covf
               + (1.0f - c2) * (1e8f + seqd * 1e6f)
               + (1.0f - rmiF[v] * rmjF) * 1e10f;
      sD[(size_t)(v + 8 * hi) * NN + j] = Da;
    }
  }
  __syncthreads();

  // ---- phase 3: per-row top-K (ascending D_adjust, index tie-break) ----
  for (int rr = 0; rr < 2; ++rr) {
    int lr = 2 * wv + rr;
    int i = r0 + lr;
    float* rowD = sD + (size_t)lr * NN;
    size_t obase = ((size_t)b * NN + i) * (size_t)K;

    for (int k = 0; k < K; ++k) {
      float best = INFINITY;
      int bi = NN;
      for (int jj = lane; jj < NN; jj += 32) {
        float v = rowD[jj];
        if (v < best) { best = v; bi = jj; }   // strict < keeps smallest index
      }
#pragma unroll
      for (int off = 16; off > 0; off >>= 1) {
        float ov = __shfl_xor(best, off, 32);
        int oi = __shfl_xor(bi, off, 32);
        if (ov < best || (ov == best && oi < bi)) { best = ov; bi = oi; }
      }
      if (lane == 0) {
        rowD[bi] = INFINITY;  // extract winner (same-wave LDS ops are in-order)
        float4 xi = sX[i], xj = sX[bi];
        float dx = xi.x - xj.x, dy = xi.y - xj.y, dz = xi.z - xj.z;
        float c2 = (float)(sCM[i] & sCM[bi]);
        float Dn = FSQRT(dx * dx + dy * dy + dz * dz + 1e-8f) * c2;
        outD[obase + k] = Dn;
        outE[obase + k] = (float)bi;
        outC[obase + k] = (Dn < 5.0e7f) ? 1.0f : 0.0f;
        outR[obase + k] = (Dn < 5.0e9f) ? 1.0f : 0.0f;
      }
    }
  }
}

// ---------------- host launcher ----------------
extern "C" void kernel_launch(void* const* d_in, const int* in_sizes, int n_in,
                              void* d_out, int out_size, void* d_ws, size_t ws_size,
                              hipStream_t stream) {
  (void)n_in; (void)d_ws; (void)ws_size; (void)in_sizes;
  const float* coords = (const float*)d_in[0];
  const unsigned char* cmask = (const unsigned char*)d_in[1];
  const int* ridx = (const int*)d_in[2];
  const unsigned char* pmask = (const unsigned char*)d_in[3];

  // Derive K from output size: per-node floats = 7 + 9 + 4K.
  int per = out_size / (BB * NN);
  int K = (per - 16) / 4;
  if (K <= 0) K = 30;

  float* o = (float*)d_out;
  float* ns = o;                                    // (B,N,7)
  float* nv = ns + (size_t)BB * NN * 7;             // (B,N,3,3)
  float* oD = nv + (size_t)BB * NN * 9;             // (B,N,K)
  float* oE = oD + (size_t)BB * NN * K;             // (B,N,K)
  float* oC = oE + (size_t)BB * NN * K;             // (B,N,K)
  float* oR = oC + (size_t)BB * NN * K;             // (B,N,K)

  // Kernel 1: node features.
  int total = BB * NN;
  gvp_node_kernel<<<(total + 255) / 256, 256, 0, stream>>>(coords, cmask, ns, nv);

  // Kernel 2: WMMA pairwise distances + top-k. One block per 16-row strip.
  dim3 grid(BB * (NN / 16));
  gvp_dist_kernel<<<grid, 256, SM_TOT, stream>>>(coords, cmask, ridx, pmask,
                                                 oD, oE, oC, oR, K);
}